// DeepSets_26963804685167
// MI455X (gfx1250) — compile-verified
//
#include <hip/hip_runtime.h>

// ---------------------------------------------------------------------------
// DeepSets on MI455X (gfx1250):
//   out[s] = (sum_{i in s} x_i) @ W + count[s] * b      (linearity of phi)
//
// Pass 1: streaming segment-sum of x (+counts) -> ws[4096][4] f32
//         (48 MB traffic => ~2.1us @ 23.3 TB/s; this is the roofline floor)
// Pass 2: 1024x V_WMMA_F32_16X16X4_F32 computing S[4096,2+1] @ [W;b] -> out
// ---------------------------------------------------------------------------

typedef __attribute__((ext_vector_type(2))) float v2f;
typedef __attribute__((ext_vector_type(8))) float v8f;

#define NSEG 4096
#define FEAT 64
#define ITER 128           // points per lane in the streaming pass
#define WAVE_PTS (32 * ITER)

// --- zero the accumulator workspace (d_ws is poisoned by the harness) ------
__global__ void ds_zero_ws(float* __restrict__ acc, int n) {
    int i = blockIdx.x * blockDim.x + threadIdx.x;
    if (i < n) acc[i] = 0.0f;
}

// --- pass 1: coalesced streaming reduce with run-length accumulation -------
// Lane l of wave w reads points  w*WAVE_PTS + i*32 + l  (contiguous 256B per
// global_load_b64 across the wave). segment_ids are sorted, so each thread
// sees long runs of equal ids (~977/32 iters per run) and only touches
// global atomics at run boundaries.
__global__ __launch_bounds__(256) void ds_seg_reduce(
        const float2* __restrict__ x,
        const int*    __restrict__ ids,
        float*        __restrict__ acc,   // [NSEG][4]: sx, sy, count, 0
        int n) {
    const int gtid = blockIdx.x * blockDim.x + threadIdx.x;
    const int wave = gtid >> 5;
    const int lane = gtid & 31;
    const int base = wave * WAVE_PTS + lane;

    int   cur = -1;
    float sx = 0.0f, sy = 0.0f, cn = 0.0f;

    #pragma unroll 4
    for (int i = 0; i < ITER; ++i) {
        const int idx = base + i * 32;
        if (idx >= n) break;
        const int id = ids[idx];
        if (id != cur) {
            if (cur >= 0) {
                atomicAdd(&acc[cur * 4 + 0], sx);
                atomicAdd(&acc[cur * 4 + 1], sy);
                atomicAdd(&acc[cur * 4 + 2], cn);
            }
            cur = id; sx = 0.0f; sy = 0.0f; cn = 0.0f;
        }
        const float2 p = x[idx];
        sx += p.x; sy += p.y; cn += 1.0f;
    }
    if (cur >= 0) {
        atomicAdd(&acc[cur * 4 + 0], sx);
        atomicAdd(&acc[cur * 4 + 1], sy);
        atomicAdd(&acc[cur * 4 + 2], cn);
    }
}

// --- pass 2: out[4096,64] = S @ [W0;W1;b;0] via V_WMMA_F32_16X16X4_F32 -----
// One wave per 16x16 output tile; 256 M-tiles x 4 N-tiles = 1024 waves.
// A (16x4 f32) layout: lanes 0-15 -> {K0,K1}, lanes 16-31 -> {K2,K3}, row=M.
// B (4x16 f32) layout: lanes 0-15 -> {K0,K1}, lanes 16-31 -> {K2,K3}, col=N.
// C/D (16x16 f32): VGPR r -> M=r (lanes 0-15) / M=8+r (lanes 16-31), N=lane%16.
__global__ __launch_bounds__(256) void ds_wmma_finish(
        const float* __restrict__ acc,   // [NSEG][4]
        const float* __restrict__ W,     // [2][64]
        const float* __restrict__ b,     // [64]
        float*       __restrict__ out) { // [NSEG][64]
    const int gtid  = blockIdx.x * blockDim.x + threadIdx.x;
    const int wid   = gtid >> 5;
    const int lane  = threadIdx.x & 31;
    const int half  = lane >> 4;         // 0: K=0,1   1: K=2,3
    const int l     = lane & 15;
    const int mtile = wid >> 2;          // 0..255
    const int ntile = wid & 3;           // 0..3

    // A operand: row M = mtile*16 + l.
    //   half==0 -> (sx, sy); half==1 -> (count, 0)  [acc[m*4+3] is zeroed]
    const int m = mtile * 16 + l;
    const float2 av = *(const float2*)(acc + m * 4 + half * 2);  // 8B aligned
    v2f a; a.x = av.x; a.y = av.y;

    // B operand: col N = ntile*16 + l.
    //   half==0 -> (W[0][n], W[1][n]); half==1 -> (b[n], 0)
    const int n = ntile * 16 + l;
    const float* brow0 = half ? b : W;              // branchless ptr select
    v2f bb;
    bb.x = brow0[n];
    bb.y = half ? 0.0f : W[FEAT + n];

    v8f c = {};
    // 8 args: (neg_a, A, neg_b, B, c_mod, C, reuse_a, reuse_b)
    c = __builtin_amdgcn_wmma_f32_16x16x4_f32(
            false, a, false, bb, (short)0, c, false, false);

    float* outp = out + (mtile * 16) * FEAT + n;
    #pragma unroll
    for (int r = 0; r < 8; ++r) {
        const int row = half ? (8 + r) : r;
        outp[row * FEAT] = c[r];
    }
}

extern "C" void kernel_launch(void* const* d_in, const int* in_sizes, int n_in,
                              void* d_out, int out_size, void* d_ws, size_t ws_size,
                              hipStream_t stream) {
    const float* x   = (const float*)d_in[0];   // [N,2] f32
    const int*   ids = (const int*)  d_in[1];   // [N]   i32 (sorted)
    const float* W   = (const float*)d_in[2];   // [2,64] f32
    const float* b   = (const float*)d_in[3];   // [64]  f32
    float*       out = (float*)d_out;           // [4096,64] f32
    float*       acc = (float*)d_ws;            // [4096][4] f32 accumulators

    const int n = in_sizes[0] / 2;              // number of points (4M)

    // zero accumulators (every call: deterministic, graph-safe)
    {
        const int elems = NSEG * 4;
        ds_zero_ws<<<(elems + 255) / 256, 256, 0, stream>>>(acc, elems);
    }
    // streaming segment reduce
    {
        const int waves  = (n + WAVE_PTS - 1) / WAVE_PTS;
        const int blocks = (waves + 7) / 8;     // 8 waves / 256-thread block
        ds_seg_reduce<<<blocks, 256, 0, stream>>>((const float2*)x, ids, acc, n);
    }
    // WMMA finisher: 1024 waves = 128 blocks of 256 threads
    {
        ds_wmma_finish<<<128, 256, 0, stream>>>(acc, W, b, out);
    }
}